// BidirectionalMamba_63067299775142
// MI455X (gfx1250) — compile-verified
//
#include <hip/hip_runtime.h>
#include <hip/hip_bf16.h>
#include <math.h>

typedef __attribute__((ext_vector_type(2))) float v2f;
typedef __attribute__((ext_vector_type(8))) float v8f;

#define D_MODEL 512
#define D_STATE 64
#define D_CONV  4
#define D_INNER 1024
#define DT_RANK 32
#define BATCH   2
#define SEQ     1024
#define ROWS    (BATCH * SEQ)           // 2048
#define XDBL_N  (DT_RANK + 2 * D_STATE) // 160

// ---------------------------------------------------------------------------
// gfx1250 async global->LDS staging (ASYNCcnt path). Fallback: VGPR bounce.
// ---------------------------------------------------------------------------
#if defined(__gfx1250__) && __has_builtin(__builtin_amdgcn_global_load_async_to_lds_b128)
#define HAVE_ASYNC_LDS 1
#else
#define HAVE_ASYNC_LDS 0
#endif

typedef int v4i_vs __attribute__((vector_size(16)));
typedef __attribute__((address_space(1))) v4i_vs* gas1_v4i;
typedef __attribute__((address_space(3))) v4i_vs* gas3_v4i;

__device__ __forceinline__ void async_copy16(const float* g, float* l) {
#if HAVE_ASYNC_LDS
    __builtin_amdgcn_global_load_async_to_lds_b128(
        (gas1_v4i)(uintptr_t)g,
        (gas3_v4i)(uintptr_t)(void*)l,
        0, 0);
#else
    (void)g; (void)l;
#endif
}

// Wait until at most the 3 newest async ops remain (previous stage done).
__device__ __forceinline__ void async_wait_prev3() {
#if HAVE_ASYNC_LDS
#if __has_builtin(__builtin_amdgcn_s_wait_asynccnt)
    __builtin_amdgcn_s_wait_asynccnt(3);
#else
    asm volatile("s_wait_asynccnt 0x3" ::: "memory");
#endif
#endif
}

__device__ __forceinline__ void async_wait_all() {
#if HAVE_ASYNC_LDS
#if __has_builtin(__builtin_amdgcn_s_wait_asynccnt)
    __builtin_amdgcn_s_wait_asynccnt(0);
#else
    asm volatile("s_wait_asynccnt 0x0" ::: "memory");
#endif
#endif
}

// ---------------------------------------------------------------------------
// Generic fp32 WMMA GEMM: C[M,N] = A[M,K] * B[K,N]  (+ C if accumulate)
// Block tile 64x128, 8 waves (2x4), each wave 2x2 grid of 16x16 WMMA tiles.
// K staged through double-buffered LDS in chunks of 16 (4 k-steps of
// V_WMMA_F32_16X16X4_F32); next stage's async copies overlap current compute.
// Per-thread async-op count is uniform (3/stage): OOB addresses are clamped
// to valid dummy rows, never skipped, so s_wait_asynccnt<=3 is exact.
// A tile rows padded to 20 floats: 16B-contiguous per async lane AND
// conflict-free column gathers (stride 20 -> 16 distinct banks / 16 lanes).
// ---------------------------------------------------------------------------
#define TBM 64
#define TBN 128
#define TBK 16
#define APAD 20

__global__ __launch_bounds__(256) void gemm_wmma_f32(
    const float* __restrict__ A, int lda,
    const float* __restrict__ B, int ldb,
    float* __restrict__ C, int ldc,
    int M, int N, int K, int accumulate)
{
    __shared__ alignas(16) float As[2][TBM][APAD];   // As[buf][m][k], padded
    __shared__ alignas(16) float Bs[2][TBK][TBN];    // Bs[buf][k][n]

    const int t      = threadIdx.x;
    const int lane   = t & 31;
    const int wave   = t >> 5;           // 0..7
    const int waveM  = wave >> 2;        // 0..1
    const int waveN  = wave & 3;         // 0..3
    const int wm     = waveM * 32;
    const int wn     = waveN * 32;
    const int half   = lane >> 4;        // 0: K pair {0,1}, 1: K pair {2,3}
    const int arow   = lane & 15;
    const int bcol   = lane & 15;

    const int blockM = blockIdx.y * TBM;
    const int blockN = blockIdx.x * TBN;

    // Static staging assignments (one A float4 + two B float4 per thread).
    const int a_m   = t >> 2;            // 0..63
    const int a_k4  = (t & 3) * 4;       // 0,4,8,12
    const int b_n4  = (t & 31) * 4;      // float4 col within tile
    const int b_k   = t >> 5;            // k row 0..7 (i=0); +8 for i=1

    // Clamped (always-valid) global coordinates; M here is always a multiple
    // of 64 and K of 16, so only the N direction can actually clamp.
    const int a_gr = (blockM + a_m < M) ? (blockM + a_m) : (M - 1);
    const int b_gc = (blockN + b_n4 + 3 < N) ? (blockN + b_n4) : 0;

    v8f acc[2][2];
    const v8f vzero = {0.f, 0.f, 0.f, 0.f, 0.f, 0.f, 0.f, 0.f};
    acc[0][0] = vzero; acc[0][1] = vzero; acc[1][0] = vzero; acc[1][1] = vzero;

    const int nStages = (K + TBK - 1) / TBK;

    // Stage a (kbase, buf) tile. Async: exactly 3 copies per thread.
    auto issue_stage = [&](int kbase, int buf) {
#if HAVE_ASYNC_LDS
        async_copy16(&A[(size_t)a_gr * lda + kbase + a_k4], &As[buf][a_m][a_k4]);
        #pragma unroll
        for (int i = 0; i < 2; ++i) {
            int k = b_k + i * 8;
            async_copy16(&B[(size_t)(kbase + k) * ldb + b_gc], &Bs[buf][k][b_n4]);
        }
#else
        #pragma unroll
        for (int i = 0; i < 4; ++i) {
            int idx = t + i * 256;
            int k = idx & 15, m = idx >> 4;
            int gr = blockM + m, gk = kbase + k;
            float v = 0.f;
            if (gr < M && gk < K) v = A[(size_t)gr * lda + gk];
            As[buf][m][k] = v;
        }
        #pragma unroll
        for (int i = 0; i < 8; ++i) {
            int idx = t + i * 256;
            int n = idx & 127, k = idx >> 7;
            int gc = blockN + n, gk = kbase + k;
            float v = 0.f;
            if (gk < K && gc < N) v = B[(size_t)gk * ldb + gc];
            Bs[buf][k][n] = v;
        }
#endif
    };

    issue_stage(0, 0);

    for (int s = 0; s < nStages; ++s) {
        const int cur = s & 1;
        if (s + 1 < nStages) {
            issue_stage((s + 1) * TBK, cur ^ 1);   // overlap with compute below
            async_wait_prev3();                    // stage s's copies complete
        } else {
            async_wait_all();
        }
        __syncthreads();

        #pragma unroll
        for (int kk = 0; kk < 4; ++kk) {
            const int kA = kk * 4 + half * 2;
            v2f afrag[2], bfrag[2];
            #pragma unroll
            for (int sm = 0; sm < 2; ++sm) {
                afrag[sm].x = As[cur][wm + sm * 16 + arow][kA];
                afrag[sm].y = As[cur][wm + sm * 16 + arow][kA + 1];
            }
            #pragma unroll
            for (int sn = 0; sn < 2; ++sn) {
                bfrag[sn].x = Bs[cur][kA][wn + sn * 16 + bcol];
                bfrag[sn].y = Bs[cur][kA + 1][wn + sn * 16 + bcol];
            }
            #pragma unroll
            for (int sm = 0; sm < 2; ++sm)
                #pragma unroll
                for (int sn = 0; sn < 2; ++sn)
                    acc[sm][sn] = __builtin_amdgcn_wmma_f32_16x16x4_f32(
                        false, afrag[sm], false, bfrag[sn],
                        (short)0, acc[sm][sn], false, false);
        }
        __syncthreads();   // all readers done before buf `cur` is overwritten
    }

    // Epilogue: D layout — VGPR i: lanes 0-15 -> M=i, lanes 16-31 -> M=i+8.
    #pragma unroll
    for (int sm = 0; sm < 2; ++sm) {
        #pragma unroll
        for (int sn = 0; sn < 2; ++sn) {
            int row0 = blockM + wm + sm * 16 + (half ? 8 : 0);
            int col  = blockN + wn + sn * 16 + bcol;
            if (col < N) {
                #pragma unroll
                for (int i = 0; i < 8; ++i) {
                    int r = row0 + i;
                    if (r < M) {
                        float v = acc[sm][sn][i];
                        if (accumulate) v += C[(size_t)r * ldc + col];
                        C[(size_t)r * ldc + col] = v;
                    }
                }
            }
        }
    }
}

// ---------------------------------------------------------------------------
// x_rev[b,l,:] = x[b, L-1-l, :]
// ---------------------------------------------------------------------------
__global__ __launch_bounds__(256) void flip_kernel(const float* __restrict__ x,
                                                   float* __restrict__ xr)
{
    int idx = blockIdx.x * blockDim.x + threadIdx.x;
    if (idx >= BATCH * SEQ * D_MODEL) return;
    int b = idx / (SEQ * D_MODEL);
    int rem = idx - b * (SEQ * D_MODEL);
    int l = rem / D_MODEL;
    int c = rem - l * D_MODEL;
    xr[idx] = x[((size_t)b * SEQ + (SEQ - 1 - l)) * D_MODEL + c];
}

__device__ __forceinline__ float silu_f(float v) {
    return v / (1.f + __expf(-v));
}

// ---------------------------------------------------------------------------
// Depthwise causal conv (width 4) + SiLU.  xi = xz[:, 0:D_INNER].
// ---------------------------------------------------------------------------
__global__ __launch_bounds__(256) void conv_silu_kernel(
    const float* __restrict__ xz, const float* __restrict__ conv_w,
    const float* __restrict__ conv_b, float* __restrict__ xc)
{
    int idx = blockIdx.x * blockDim.x + threadIdx.x;
    if (idx >= BATCH * SEQ * D_INNER) return;
    int b = idx / (SEQ * D_INNER);
    int rem = idx - b * (SEQ * D_INNER);
    int l = rem / D_INNER;
    int d = rem - l * D_INNER;
    float sum = conv_b[d];
    #pragma unroll
    for (int j = 0; j < D_CONV; ++j) {
        int ll = l - (D_CONV - 1) + j;
        if (ll >= 0)
            sum += conv_w[d * D_CONV + j] *
                   xz[((size_t)b * SEQ + ll) * (2 * D_INNER) + d];
    }
    xc[idx] = silu_f(sum);
}

// ---------------------------------------------------------------------------
// dt[i] = softplus(dt_pre[i] + b_dt[i % D_INNER])   (in place)
// ---------------------------------------------------------------------------
__global__ __launch_bounds__(256) void softplus_kernel(float* __restrict__ dt,
                                                       const float* __restrict__ b_dt)
{
    int idx = blockIdx.x * blockDim.x + threadIdx.x;
    if (idx >= ROWS * D_INNER) return;
    float v = dt[idx] + b_dt[idx & (D_INNER - 1)];
    dt[idx] = (v > 20.f) ? v : log1pf(__expf(v));
}

// ---------------------------------------------------------------------------
// Selective scan, one wave per (b,d). Lane holds states n and n+32 in regs.
// Software-pipelined: loads for step l+1 issue while step l computes; rows
// 16 steps ahead are prefetched into L2/WGP$ (global_prefetch_b8).
// Fuses y = h.C + D*xc, g = 0.5 * y * silu(z), and backward output flip.
// ---------------------------------------------------------------------------
#define SCAN_PF 16

__global__ __launch_bounds__(256) void scan_kernel(
    const float* __restrict__ dt, const float* __restrict__ xc,
    const float* __restrict__ xdbl, const float* __restrict__ A_log,
    const float* __restrict__ Dp, const float* __restrict__ xz,
    float* __restrict__ g, int flip)
{
    int wid = (blockIdx.x * blockDim.x + threadIdx.x) >> 5;
    int lane = threadIdx.x & 31;
    if (wid >= BATCH * D_INNER) return;
    int b = wid >> 10;           // / D_INNER
    int d = wid & (D_INNER - 1);

    const int n1 = lane, n2 = lane + 32;
    const float A0 = -__expf(A_log[d * D_STATE + n1]);
    const float A1 = -__expf(A_log[d * D_STATE + n2]);
    const float Dv = Dp[d];
    const int r0 = b * SEQ;

    // Pipeline registers: data for the step about to be computed.
    float dtv = dt[(size_t)r0 * D_INNER + d];
    float xv  = xc[(size_t)r0 * D_INNER + d];
    const float* xd0 = xdbl + (size_t)r0 * XDBL_N;
    float B0 = xd0[DT_RANK + n1];
    float B1 = xd0[DT_RANK + n2];
    float C0 = xd0[DT_RANK + D_STATE + n1];
    float C1 = xd0[DT_RANK + D_STATE + n2];

    float h0 = 0.f, h1 = 0.f;
    for (int l = 0; l < SEQ; ++l) {
        const int r = r0 + l;

        // Prefetch far-ahead rows (one cacheline-granular touch per lane).
        if (l + SCAN_PF < SEQ) {
            const int rp = r + SCAN_PF;
            __builtin_prefetch((const char*)(xdbl + (size_t)rp * XDBL_N) + lane * 32, 0, 0);
            __builtin_prefetch(&dt[(size_t)rp * D_INNER + d], 0, 0);
            __builtin_prefetch(&xc[(size_t)rp * D_INNER + d], 0, 0);
        }

        // Issue next step's loads (independent of the recurrence).
        float ndt = 0.f, nxv = 0.f, nB0 = 0.f, nB1 = 0.f, nC0 = 0.f, nC1 = 0.f;
        if (l + 1 < SEQ) {
            const int rn = r + 1;
            ndt = dt[(size_t)rn * D_INNER + d];
            nxv = xc[(size_t)rn * D_INNER + d];
            const float* xdn = xdbl + (size_t)rn * XDBL_N;
            nB0 = xdn[DT_RANK + n1];
            nB1 = xdn[DT_RANK + n2];
            nC0 = xdn[DT_RANK + D_STATE + n1];
            nC1 = xdn[DT_RANK + D_STATE + n2];
        }

        // Recurrence + output contraction for step l.
        const float u = dtv * xv;
        h0 = __expf(dtv * A0) * h0 + u * B0;
        h1 = __expf(dtv * A1) * h1 + u * B1;
        float p = h0 * C0 + h1 * C1;
        #pragma unroll
        for (int off = 16; off > 0; off >>= 1)
            p += __shfl_xor(p, off, 32);
        if (lane == 0) {
            float y = p + Dv * xv;
            float zv = xz[(size_t)r * (2 * D_INNER) + D_INNER + d];
            float gv = 0.5f * y * silu_f(zv);
            int lo = flip ? (SEQ - 1 - l) : l;
            g[((size_t)b * SEQ + lo) * D_INNER + d] = gv;
        }

        dtv = ndt; xv = nxv; B0 = nB0; B1 = nB1; C0 = nC0; C1 = nC1;
    }
}

// ---------------------------------------------------------------------------
extern "C" void kernel_launch(void* const* d_in, const int* in_sizes, int n_in,
                              void* d_out, int out_size, void* d_ws, size_t ws_size,
                              hipStream_t stream)
{
    const float* x = (const float*)d_in[0];
    // f_* params: indices 1..9, b_* params: 10..18
    const float* W_in[2]   = {(const float*)d_in[1],  (const float*)d_in[10]};
    const float* conv_w[2] = {(const float*)d_in[2],  (const float*)d_in[11]};
    const float* conv_b[2] = {(const float*)d_in[3],  (const float*)d_in[12]};
    const float* W_x[2]    = {(const float*)d_in[4],  (const float*)d_in[13]};
    const float* W_dt[2]   = {(const float*)d_in[5],  (const float*)d_in[14]};
    const float* b_dt[2]   = {(const float*)d_in[6],  (const float*)d_in[15]};
    const float* A_log[2]  = {(const float*)d_in[7],  (const float*)d_in[16]};
    const float* Dp[2]     = {(const float*)d_in[8],  (const float*)d_in[17]};
    const float* W_out[2]  = {(const float*)d_in[9],  (const float*)d_in[18]};
    float* out = (float*)d_out;

    float* ws = (float*)d_ws;
    size_t off = 0;
    float* x_rev = ws + off; off += (size_t)ROWS * D_MODEL;
    float* xzbuf = ws + off; off += (size_t)ROWS * 2 * D_INNER;
    float* xcbuf = ws + off; off += (size_t)ROWS * D_INNER;
    float* xdbl  = ws + off; off += (size_t)ROWS * XDBL_N;
    float* dtbuf = ws + off; off += (size_t)ROWS * D_INNER;
    float* gbuf[2];
    gbuf[0] = ws + off; off += (size_t)ROWS * D_INNER;
    gbuf[1] = ws + off; off += (size_t)ROWS * D_INNER;

    // Flip input for the backward direction.
    flip_kernel<<<(ROWS * D_MODEL + 255) / 256, 256, 0, stream>>>(x, x_rev);

    for (int dir = 0; dir < 2; ++dir) {
        const float* xin = (dir == 0) ? x : x_rev;

        // xz = xin @ W_in   (2048 x 512 x 2048)
        {
            dim3 grid((2 * D_INNER + TBN - 1) / TBN, (ROWS + TBM - 1) / TBM);
            gemm_wmma_f32<<<grid, 256, 0, stream>>>(
                xin, D_MODEL, W_in[dir], 2 * D_INNER, xzbuf, 2 * D_INNER,
                ROWS, 2 * D_INNER, D_MODEL, 0);
        }

        // xc = silu(causal_conv(xi) + conv_b)
        conv_silu_kernel<<<(ROWS * D_INNER + 255) / 256, 256, 0, stream>>>(
            xzbuf, conv_w[dir], conv_b[dir], xcbuf);

        // x_dbl = xc @ W_x   (2048 x 1024 x 160)
        {
            dim3 grid((XDBL_N + TBN - 1) / TBN, (ROWS + TBM - 1) / TBM);
            gemm_wmma_f32<<<grid, 256, 0, stream>>>(
                xcbuf, D_INNER, W_x[dir], XDBL_N, xdbl, XDBL_N,
                ROWS, XDBL_N, D_INNER, 0);
        }

        // dt_pre = dt_raw @ W_dt   (2048 x 32 x 1024); dt_raw = x_dbl[:, :32]
        {
            dim3 grid((D_INNER + TBN - 1) / TBN, (ROWS + TBM - 1) / TBM);
            gemm_wmma_f32<<<grid, 256, 0, stream>>>(
                xdbl, XDBL_N, W_dt[dir], D_INNER, dtbuf, D_INNER,
                ROWS, D_INNER, DT_RANK, 0);
        }

        // dt = softplus(dt_pre + b_dt)
        softplus_kernel<<<(ROWS * D_INNER + 255) / 256, 256, 0, stream>>>(
            dtbuf, b_dt[dir]);

        // Selective scan + fused gating/scale/flip -> g
        scan_kernel<<<(BATCH * D_INNER * 32 + 255) / 256, 256, 0, stream>>>(
            dtbuf, xcbuf, xdbl, A_log[dir], Dp[dir], xzbuf, gbuf[dir], dir);
    }

    // out = g_f @ f_W_out + g_b @ b_W_out   (0.5 folded into g)
    {
        dim3 grid((D_MODEL + TBN - 1) / TBN, (ROWS + TBM - 1) / TBM);
        gemm_wmma_f32<<<grid, 256, 0, stream>>>(
            gbuf[0], D_INNER, W_out[0], D_MODEL, out, D_MODEL,
            ROWS, D_MODEL, D_INNER, 0);
        gemm_wmma_f32<<<grid, 256, 0, stream>>>(
            gbuf[1], D_INNER, W_out[1], D_MODEL, out, D_MODEL,
            ROWS, D_MODEL, D_INNER, 1);
    }
}